// Box3DHeuristic_738734375365
// MI455X (gfx1250) — compile-verified
//
#include <hip/hip_runtime.h>
#include <math.h>

typedef __attribute__((ext_vector_type(2))) float v2f;
typedef __attribute__((ext_vector_type(8))) float v8f;

#define NDIR 800
#define WPB  8      // waves per block (wave32)
#define GSTR 34     // padded G-tile row stride (conflict-free, 8B aligned)

__global__ __launch_bounds__(256)
void ffi_moment_kernel(const float* __restrict__ F4_in,
                       const float* __restrict__ dir_x,
                       const float* __restrict__ dir_y,
                       const float* __restrict__ dir_z,
                       const float* __restrict__ quad_w,
                       float* __restrict__ outF4,
                       float* __restrict__ outG,
                       int Btot)
{
  constexpr float EPS    = 1e-12f;
  constexpr float INV4PI = 0.07957747154594767f;

  __shared__ float sDIR[NDIR][4];       // dx, dy, dz, w
  __shared__ float sBW [NDIR][4];       // w, w*dx, w*dy, w*dz   (WMMA B source)
  __shared__ float sG  [WPB][16][GSTR]; // per-wave A tile: 16 rows x 32 dirs
  __shared__ float sS  [WPB][16][16];   // per-wave D-matrix dump

  const int tid  = threadIdx.x;
  const int wave = tid >> 5;
  const int lane = tid & 31;

  for (int i = tid; i < NDIR; i += 256) {
    float dx = dir_x[i], dy = dir_y[i], dz = dir_z[i], w = quad_w[i];
    sDIR[i][0] = dx; sDIR[i][1] = dy; sDIR[i][2] = dz; sDIR[i][3] = w;
    sBW [i][0] = w;  sBW [i][1] = w*dx; sBW[i][2] = w*dy; sBW[i][3] = w*dz;
  }
  __syncthreads();   // only true cross-wave dependency in the kernel

  const int pair = blockIdx.x * WPB + wave;   // 2 batches per wave
  int bidx[2];
  bidx[0] = min(pair*2,     Btot-1);
  bidx[1] = min(pair*2 + 1, Btot-1);

  // ---- per-batch uniform setup: N, F, fluxfac -> Z, C = N/(4pi)*z/sinh(z), unit flux dir
  float Cc[2][4], Zc[2][4], Ax[2][4], Ay[2][4], Az[2][4];
#pragma unroll
  for (int bt = 0; bt < 2; ++bt) {
    const float* p = F4_in + (size_t)bidx[bt] * 24;
    float Ns[4], Fx[4], Fy[4], Fz[4];
    // species order: e, ebar, x, xbar
    Ns[0] = p[3];               Fx[0] = p[0];               Fy[0] = p[1];               Fz[0] = p[2];
    Ns[1] = p[15];              Fx[1] = p[12];              Fy[1] = p[13];              Fz[1] = p[14];
    Ns[2] = 0.5f*(p[7]+p[11]);  Fx[2] = 0.5f*(p[4]+p[8]);   Fy[2] = 0.5f*(p[5]+p[9]);   Fz[2] = 0.5f*(p[6]+p[10]);
    Ns[3] = 0.5f*(p[19]+p[23]); Fx[3] = 0.5f*(p[16]+p[20]); Fy[3] = 0.5f*(p[17]+p[21]); Fz[3] = 0.5f*(p[18]+p[22]);
#pragma unroll
    for (int s = 0; s < 4; ++s) {
      float N     = fmaxf(Ns[s], EPS);
      float nrm2  = Fx[s]*Fx[s] + Fy[s]*Fy[s] + Fz[s]*Fz[s];
      float normF = sqrtf(nrm2);
      float f  = fminf(fmaxf(normF / N, 0.0f), 0.999999f);
      float f2 = f*f, f4 = f2*f2, f6 = f4*f2, f8 = f4*f4;
      float denom = 3.0f - 1.00651f*f2 - 0.962251f*f4 + 1.47353f*f6 - 0.48953f*f8;
      float pz = 1.0f - 2.0f*(1.0f - f)*(1.0f + 1.01524f*f) / fmaxf(denom, EPS);
      float z  = 2.0f*f / fmaxf(1.0f - pz, 1e-6f);
      bool  small  = fabsf(z) < 1e-4f;
      float zs     = small ? 1.0f : z;
      float z2     = z*z;
      float series = 1.0f - z2*(1.0f/6.0f) + z2*z2*(1.0f/120.0f);
      float sh     = 0.5f*(__expf(zs) - __expf(-zs));
      float zz     = small ? series : (zs / sh);
      Cc[bt][s] = N * INV4PI * zz;
      Zc[bt][s] = z;
      float inorm = 1.0f / fmaxf(normF, EPS);
      Ax[bt][s] = Fx[s]*inorm; Ay[bt][s] = Fy[s]*inorm; Az[bt][s] = Fz[s]*inorm;
    }
  }

  float Ip0 = 0.f, Im0 = 0.f, Ip1 = 0.f, Im1 = 0.f;
  v8f acc = {0.f,0.f,0.f,0.f,0.f,0.f,0.f,0.f};

  const int row16 = lane & 15;           // WMMA row (A) / col (B) for this lane
  const int khalf = (lane >> 4) << 1;    // 0 for lanes 0-15, 2 for lanes 16-31
  const int bc    = row16 & 3;           // clamped B column (cols 4..15 are zero)
  const float bmask = (row16 < 4) ? 1.0f : 0.0f;

  // ---- main loop: 25 chunks of 32 directions.
  // sG is PER-WAVE private: same-wave DS ops are processed in order by the LDS
  // unit (CDNA5 ISA 7.3), so only a compiler scheduling fence is needed between
  // this wave's stores and its own cross-lane loads -- no workgroup barrier.
  for (int it = 0; it < NDIR/32; ++it) {
    const int d = it*32 + lane;
    const float dx = sDIR[d][0], dy = sDIR[d][1], dz = sDIR[d][2], w = sDIR[d][3];

#pragma unroll
    for (int bt = 0; bt < 2; ++bt) {
      float g[4];
#pragma unroll
      for (int s = 0; s < 4; ++s) {
        float mu = Ax[bt][s]*dx + Ay[bt][s]*dy + Az[bt][s]*dz;
        g[s] = Cc[bt][s] * __expf(Zc[bt][s] * mu);
      }
      float delta = g[0] - g[1] - g[2] + g[3];
      float dpos  = fmaxf(delta, 0.f);
      float dneg  = fmaxf(-delta, 0.f);
      if (bt == 0) { Ip0 += w*dpos; Im0 += w*dneg; }
      else         { Ip1 += w*dpos; Im1 += w*dneg; }
      bool  pos = (delta >= 0.f);
      float dex = g[0] - g[2];
      float deb = g[1] - g[3];
      const int rb = bt*8;
      sG[wave][rb+0][lane] = g[0];
      sG[wave][rb+1][lane] = g[1];
      sG[wave][rb+2][lane] = g[2];
      sG[wave][rb+3][lane] = g[3];
      sG[wave][rb+4][lane] = pos ? dex : 0.f;
      sG[wave][rb+5][lane] = pos ? 0.f : dex;
      sG[wave][rb+6][lane] = pos ? deb : 0.f;
      sG[wave][rb+7][lane] = pos ? 0.f : deb;
    }
    __builtin_amdgcn_wave_barrier();   // fence: stores above vs lane-crossing loads below

    // 8 WMMAs cover K = 32 directions (K=4 each). M=16 rows = 2 batches x 8
    // generalized densities, N cols 0..3 = moments (w, w*dx, w*dy, w*dz).
#pragma unroll
    for (int k4 = 0; k4 < 8; ++k4) {
      const int kl = k4*4 + khalf;      // local K index within tile
      const int kd = it*32 + kl;        // global direction index
      v2f a, b;
      a.x = sG[wave][row16][kl];
      a.y = sG[wave][row16][kl+1];
      b.x = bmask * sBW[kd][bc];        // unconditional load + mask: EXEC untouched
      b.y = bmask * sBW[kd+1][bc];
      acc = __builtin_amdgcn_wmma_f32_16x16x4_f32(
              /*neg_a=*/false, a, /*neg_b=*/false, b,
              /*c_mod=*/(short)0, acc, /*reuse_a=*/false, /*reuse_b=*/false);
    }
    __builtin_amdgcn_wave_barrier();   // fence: loads above vs next chunk's stores
  }

  // ---- wave-level reduction of I+/I- (wave32)
#pragma unroll
  for (int off = 16; off > 0; off >>= 1) {
    Ip0 += __shfl_xor(Ip0, off, 32);
    Im0 += __shfl_xor(Im0, off, 32);
    Ip1 += __shfl_xor(Ip1, off, 32);
    Im1 += __shfl_xor(Im1, off, 32);
  }

  // ---- dump WMMA accumulator (16x16 f32 C/D layout) to per-wave LDS
  const int mhi = (lane >= 16) ? 8 : 0;
#pragma unroll
  for (int j = 0; j < 8; ++j)
    sS[wave][j + mhi][row16] = acc[j];
  __builtin_amdgcn_wave_barrier();     // per-wave private dump; same-wave DS in order

  // ---- final combine: one lane per batch
  if (lane < 2) {
    const int bt = lane;
    const int b  = pair*2 + bt;
    if (b < Btot) {
      const float ipv = (bt == 0) ? Ip0 : Ip1;
      const float imv = (bt == 0) ? Im0 : Im1;
      const float ips = fmaxf(ipv, EPS);
      const float ims = fmaxf(imv, EPS);
      float cp = (ipv < imv) ? (1.0f/3.0f) : (1.0f - (2.0f/3.0f)*(ims/ips));
      float cn = (ipv < imv) ? (1.0f - (2.0f/3.0f)*(ips/ims)) : (1.0f/3.0f);
      cp = fminf(fmaxf(cp, 0.f), 1.f);
      cn = fminf(fmaxf(cn, 0.f), 1.f);

      float Me[4], Meb[4], Mx[4], Mxb[4];
#pragma unroll
      for (int c = 0; c < 4; ++c) {
        const float Se  = sS[wave][bt*8+0][c];
        const float Seb = sS[wave][bt*8+1][c];
        const float Sx  = sS[wave][bt*8+2][c];
        const float Sxb = sS[wave][bt*8+3][c];
        const float mE  = cp*sS[wave][bt*8+4][c] + cn*sS[wave][bt*8+5][c];
        const float mB  = cp*sS[wave][bt*8+6][c] + cn*sS[wave][bt*8+7][c];
        Me [c] = Sx  + mE;
        Meb[c] = Sxb + mB;
        Mx [c] = 0.5f*(Se  + Sx ) - 0.5f*mE;
        Mxb[c] = 0.5f*(Seb + Sxb) - 0.5f*mB;
      }

      float* o = outF4 + (size_t)b * 24;
      // (b, 0, 0, :) = nu_e ; (b, 0, 1:3, :) = nu_x
      o[0]  = Me[1];  o[1]  = Me[2];  o[2]  = Me[3];  o[3]  = Me[0];
      o[4]  = Mx[1];  o[5]  = Mx[2];  o[6]  = Mx[3];  o[7]  = Mx[0];
      o[8]  = Mx[1];  o[9]  = Mx[2];  o[10] = Mx[3];  o[11] = Mx[0];
      // (b, 1, 0, :) = nubar_e ; (b, 1, 1:3, :) = nubar_x
      o[12] = Meb[1]; o[13] = Meb[2]; o[14] = Meb[3]; o[15] = Meb[0];
      o[16] = Mxb[1]; o[17] = Mxb[2]; o[18] = Mxb[3]; o[19] = Mxb[0];
      o[20] = Mxb[1]; o[21] = Mxb[2]; o[22] = Mxb[3]; o[23] = Mxb[0];

      outG[b] = sqrtf(fmaxf(ipv * imv, 0.f));
    }
  }
}

extern "C" void kernel_launch(void* const* d_in, const int* in_sizes, int n_in,
                              void* d_out, int out_size, void* d_ws, size_t ws_size,
                              hipStream_t stream) {
  const float* F4 = (const float*)d_in[0];
  const float* dx = (const float*)d_in[1];
  const float* dy = (const float*)d_in[2];
  const float* dz = (const float*)d_in[3];
  const float* qw = (const float*)d_in[4];
  const int B = in_sizes[0] / 24;           // (B, 2, 3, 4) f32
  float* outF4 = (float*)d_out;             // B*24 floats
  float* outG  = outF4 + (size_t)B * 24;    // B floats (growth)
  const int batchesPerBlock = 2 * WPB;      // 16
  const int grid = (B + batchesPerBlock - 1) / batchesPerBlock;
  hipLaunchKernelGGL(ffi_moment_kernel, dim3(grid), dim3(256), 0, stream,
                     F4, dx, dy, dz, qw, outF4, outG, B);
}